// MVDR_19026705121950
// MI455X (gfx1250) — compile-verified
//
#include <hip/hip_runtime.h>

typedef __attribute__((ext_vector_type(2))) float v2f;
typedef __attribute__((ext_vector_type(8))) float v8f;
typedef __attribute__((ext_vector_type(4))) unsigned int u32x4;
typedef __attribute__((ext_vector_type(8))) int i32x8;
typedef __attribute__((ext_vector_type(4))) int i32x4;

#define B_ 8
#define C_ 6
#define F_ 257
#define T_ 2000

__device__ __forceinline__ float2 cmul(float2 a, float2 b) {
    return make_float2(a.x * b.x - a.y * b.y, a.x * b.y + a.y * b.x);
}
__device__ __forceinline__ float2 cinv(float2 a) {
    float r = 1.0f / (a.x * a.x + a.y * a.y);
    return make_float2(a.x * r, -a.y * r);
}

// Flat shared pointer -> LDS byte offset (aperture maps LDS addr = flat[31:0])
__device__ __forceinline__ unsigned lds_off(const void* p) {
    return (unsigned)(size_t)p;
}

// Issue one TDM 2D tile load: rows x cols f32, row stride in elements.
// D# layout per CDNA5 ISA.
__device__ __forceinline__ void tdm_load_2d(unsigned lds_byte, const void* gptr,
                                            unsigned rows, unsigned cols,
                                            unsigned row_stride_elems) {
    unsigned long long ga = (unsigned long long)(size_t)gptr;
    u32x4 g0;
    g0[0] = 1u;                                            // count=1 (valid user D#)
    g0[1] = lds_byte;                                      // lds_addr
    g0[2] = (unsigned)(ga & 0xFFFFFFFFu);                  // global_addr[31:0]
    g0[3] = (unsigned)((ga >> 32) & 0x01FFFFFFu)           // global_addr[56:32]
          | (2u << 30);                                    // type=2 ("image")
    i32x8 g1;
    g1[0] = (int)(2u << 16);                               // data_size=2 -> 4 bytes
    g1[1] = (int)((cols & 0xFFFFu) << 16);                 // tensor_dim0[15:0] @bit48
    g1[2] = (int)((cols >> 16) | ((rows & 0xFFFFu) << 16));// tensor_dim0 hi | tensor_dim1 lo
    g1[3] = (int)((rows >> 16) | (cols << 16));            // tensor_dim1 hi | tile_dim0
    g1[4] = (int)(rows & 0xFFFFu);                         // tile_dim1 | tile_dim2=0
    g1[5] = (int)row_stride_elems;                         // tensor_dim0_stride[31:0]
    g1[6] = 0;                                             // stride hi | dim1_stride lo
    g1[7] = 0;                                             // dim1_stride hi
    i32x4 z4 = {0, 0, 0, 0};
    i32x8 z8 = {0, 0, 0, 0, 0, 0, 0, 0};
    __builtin_amdgcn_tensor_load_to_lds(g0, g1, z4, z4, z8, 0);
}

__global__ void __launch_bounds__(128) mvdr_kernel(
    const float* __restrict__ spec_re, const float* __restrict__ spec_im,
    const float* __restrict__ mask_s, const float* __restrict__ mask_n,
    float* __restrict__ out)
{
    __shared__ float lds_spec[12 * T_];  // rows 0-5: Re(ch), rows 6-11: Im(ch)  (94 KB)
    __shared__ float lds_mask[2 * T_];   // row 0: mask_s, row 1: mask_n         (16 KB)
    __shared__ float ldsD[512];          // [0:256) = D_s tile, [256:512) = D_n tile
    __shared__ float ldsSum[2];          // mask sums
    __shared__ float ldsW[12];           // beamformer weights (6 complex)

    const int bf    = blockIdx.x;
    const int b     = bf / F_;
    const int f     = bf % F_;
    const int tid   = threadIdx.x;
    const int wid   = tid >> 5;
    const int lane  = tid & 31;
    const int r     = lane & 15;          // A row / B col index
    const int khalf = lane >> 4;          // which K-pair this half-wave holds

    const size_t spec_base = ((size_t)(b * C_) * F_ + f) * (size_t)T_;

    // ---- TDM: bulk-copy spec tile + masks into LDS (wave 0 issues) ----
    if (wid == 0) {
        tdm_load_2d(lds_off(lds_spec),              spec_re + spec_base, 6, T_, F_ * T_);
        tdm_load_2d(lds_off(lds_spec) + 6 * T_ * 4, spec_im + spec_base, 6, T_, F_ * T_);
        tdm_load_2d(lds_off(lds_mask),              mask_s + (size_t)(b * F_ + f) * T_, 1, T_, T_);
        tdm_load_2d(lds_off(lds_mask) + T_ * 4,     mask_n + (size_t)(b * F_ + f) * T_, 1, T_, T_);
    }
    // zero the shared accumulators while the TDM runs
    for (int i = tid; i < 512; i += 128) ldsD[i] = 0.f;
    if (tid < 2) ldsSum[tid] = 0.f;
    if (wid == 0) __builtin_amdgcn_s_wait_tensorcnt(0);
    __syncthreads();

    // ---- mask sums (for PSD normalization; only needed at solve time) ----
    {
        float ssum = 0.f, nsum = 0.f;
        for (int t = tid; t < T_; t += 128) { ssum += lds_mask[t]; nsum += lds_mask[T_ + t]; }
        for (int o = 16; o > 0; o >>= 1) {
            ssum += __shfl_xor(ssum, o, 32);
            nsum += __shfl_xor(nsum, o, 32);
        }
        if (lane == 0) { atomicAdd(&ldsSum[0], ssum); atomicAdd(&ldsSum[1], nsum); }
    }

    // ---- PSD accumulation via V_WMMA_F32_16X16X4_F32, 4 waves split T ----
    // A rows 0..5 = m*Re, rows 6..11 = m*Im; B = raw spec (same per-lane values).
    const int r12 = (r < 12) ? r : 0;
    const float2* lsp2 = (const float2*)lds_spec;
    const float2* lmk2 = (const float2*)lds_mask;

    v8f acc_s = {0.f, 0.f, 0.f, 0.f, 0.f, 0.f, 0.f, 0.f};
    v8f acc_n = {0.f, 0.f, 0.f, 0.f, 0.f, 0.f, 0.f, 0.f};

    const int CH = (T_ / 4) / 4;          // 125 K-chunks per wave
    const int i0 = wid * CH;
    #pragma unroll 2
    for (int i = 0; i < CH; ++i) {
        const int h = 2 * (i0 + i) + khalf;        // float2 index; t0 = 2*h
        float2 v = lsp2[r12 * (T_ / 2) + h];
        float2 m = lmk2[h];
        float2 n = lmk2[T_ / 2 + h];
        v2f bfrag = {v.x, v.y};
        v2f afs   = {v.x * m.x, v.y * m.y};
        v2f afn   = {v.x * n.x, v.y * n.y};
        acc_s = __builtin_amdgcn_wmma_f32_16x16x4_f32(
            false, afs, false, bfrag, (short)0, acc_s, false, false);
        acc_n = __builtin_amdgcn_wmma_f32_16x16x4_f32(
            false, afn, false, bfrag, (short)0, acc_n, false, false);
    }

    // combine the 4 per-wave partial tiles (lane L, vgpr v -> D[v+8*khalf][r])
    #pragma unroll
    for (int v = 0; v < 8; ++v) {
        atomicAdd(&ldsD[(v + 8 * khalf) * 16 + r], acc_s[v]);
        atomicAdd(&ldsD[256 + (v + 8 * khalf) * 16 + r], acc_n[v]);
    }
    __syncthreads();

    // ---- 6x6 complex solve on wave 0 ----
    if (wid == 0) {
        const float inv_s = 1.0f / (ldsSum[0] + 1e-15f);
        const float inv_n = 1.0f / (ldsSum[1] + 1e-15f);
        const float* Ds = ldsD;
        const float* Dn = ldsD + 256;

        const int c = (lane < 6) ? lane : 5;
        float2 row[12];   // augmented [psd_n | psd_s], lane c owns row c
        #pragma unroll
        for (int e = 0; e < 6; ++e) {
            float nr = inv_n * (Dn[c * 16 + e] + Dn[(c + 6) * 16 + e + 6]);
            float ni = inv_n * (Dn[(c + 6) * 16 + e] - Dn[c * 16 + e + 6]);
            float sr = inv_s * (Ds[c * 16 + e] + Ds[(c + 6) * 16 + e + 6]);
            float si = inv_s * (Ds[(c + 6) * 16 + e] - Ds[c * 16 + e + 6]);
            row[e]     = make_float2(nr, ni);
            row[6 + e] = make_float2(sr, si);
        }

        // diagonal loading: eps = trace(psd_n).re * 1e-7 + 1e-8
        float diag = 0.f;
        #pragma unroll
        for (int e = 0; e < 6; ++e) if (lane == e) diag = row[e].x;
        float tr_n = (lane < 6) ? diag : 0.f;
        for (int o = 16; o > 0; o >>= 1) tr_n += __shfl_xor(tr_n, o, 32);
        const float epsl = tr_n * 1e-7f + 1e-8f;
        #pragma unroll
        for (int e = 0; e < 6; ++e) if (lane == e) row[e].x += epsl;

        // Gauss-Jordan (HPD, no pivoting): right half -> N^{-1} S
        #pragma unroll
        for (int k = 0; k < 6; ++k) {
            float2 pk[12];
            #pragma unroll
            for (int j = 0; j < 12; ++j) {
                pk[j].x = __shfl(row[j].x, k, 32);
                pk[j].y = __shfl(row[j].y, k, 32);
            }
            float2 pinv = cinv(pk[k]);
            if (lane == k) {
                #pragma unroll
                for (int j = 0; j < 12; ++j) row[j] = cmul(pk[j], pinv);
            } else {
                float2 fac = cmul(row[k], pinv);
                #pragma unroll
                for (int j = 0; j < 12; ++j) {
                    float2 t = cmul(fac, pk[j]);
                    row[j] = make_float2(row[j].x - t.x, row[j].y - t.y);
                }
            }
        }

        // trace(numerator); w = numerator[:, ref=0] / (tr + 1e-8)
        float2 d2 = make_float2(0.f, 0.f);
        #pragma unroll
        for (int e = 0; e < 6; ++e) if (lane == e) d2 = row[6 + e];
        float trx = (lane < 6) ? d2.x : 0.f;
        float trym = (lane < 6) ? d2.y : 0.f;
        for (int o = 16; o > 0; o >>= 1) {
            trx  += __shfl_xor(trx, o, 32);
            trym += __shfl_xor(trym, o, 32);
        }
        float2 denom = make_float2(trx + 1e-8f, trym);
        float2 w = cmul(row[6], cinv(denom));
        if (lane < 6) { ldsW[2 * lane] = w.x; ldsW[2 * lane + 1] = w.y; }
    }
    __syncthreads();

    // ---- apply beamformer from LDS-resident spec, 128 threads ----
    float wr[6], wi[6];
    #pragma unroll
    for (int e = 0; e < 6; ++e) { wr[e] = ldsW[2 * e]; wi[e] = ldsW[2 * e + 1]; }

    float2* out2 = (float2*)out + (size_t)(b * F_ + f) * T_;
    for (int t = tid; t < T_; t += 128) {
        float re = 0.f, im = 0.f;
        #pragma unroll
        for (int e = 0; e < 6; ++e) {
            float sr = lds_spec[e * T_ + t];
            float si = lds_spec[(e + 6) * T_ + t];
            re += wr[e] * sr + wi[e] * si;      // conj(w) * s
            im += wr[e] * si - wi[e] * sr;
        }
        out2[t] = make_float2(re, im);
    }
}

extern "C" void kernel_launch(void* const* d_in, const int* in_sizes, int n_in,
                              void* d_out, int out_size, void* d_ws, size_t ws_size,
                              hipStream_t stream) {
    const float* spec_re = (const float*)d_in[0];
    const float* spec_im = (const float*)d_in[1];
    const float* mask_s  = (const float*)d_in[2];
    const float* mask_n  = (const float*)d_in[3];
    float* out = (float*)d_out;

    dim3 grid(B_ * F_);
    dim3 block(128);
    mvdr_kernel<<<grid, block, 0, stream>>>(spec_re, spec_im, mask_s, mask_n, out);
}